// RGCN_2765958939332
// MI455X (gfx1250) — compile-verified
//
#include <hip/hip_runtime.h>

// ---------------------------------------------------------------------------
// R-GCN (HeteroGraphConv, sum aggregate, GraphConv norm='both') on gfx1250.
// Per layer: out = sum_r [ norm_d_r ⊙ scatter_dst( (norm_s_r ⊙ h) @ W_r ) + b_r ]
// Dense transform uses V_WMMA_F32_16X16X4_F32 (fp32, matches reference).
// Scatter uses native GLOBAL_ATOMIC_ADD_F32 via inline asm (guaranteed single-
// instruction fp32 atomic, no CAS-loop fallback); norm_d folded per edge.
// ---------------------------------------------------------------------------

typedef float v2f __attribute__((ext_vector_type(2)));
typedef float v8f __attribute__((ext_vector_type(8)));

static constexpr int R_REL = 4;
static constexpr int DFEAT = 128;   // din for every layer (128)

// Hardware fp32 atomic add, no return (tracked on STOREcnt; S_ENDPGM's
// implicit wait-idle drains it before the next stream-ordered kernel).
__device__ __forceinline__ void atomAddF32(float* p, float v) {
  asm volatile("global_atomic_add_f32 %0, %1, off"
               :
               : "v"(p), "v"(v)
               : "memory");
}

// ---------------------------------------------------------------- utilities
__global__ void __launch_bounds__(256) fill_zero_f32(float* p, size_t n) {
  size_t i = (size_t)blockIdx.x * blockDim.x + threadIdx.x;
  if (i < n) p[i] = 0.0f;
}

// grid covers R*E threads; src/dst are flattened [R,E]
__global__ void __launch_bounds__(256)
count_degrees(const int* __restrict__ src, const int* __restrict__ dst,
              float* __restrict__ degS, float* __restrict__ degD,
              int nNodes, int nEdges) {
  size_t i = (size_t)blockIdx.x * blockDim.x + threadIdx.x;
  if (i >= (size_t)R_REL * nEdges) return;
  int r = (int)(i / nEdges);
  int s = src[i];
  int d = dst[i];
  atomAddF32(&degS[(size_t)r * nNodes + s], 1.0f);
  atomAddF32(&degD[(size_t)r * nNodes + d], 1.0f);
}

__global__ void __launch_bounds__(256) deg_to_norm(float* p, size_t n) {
  size_t i = (size_t)blockIdx.x * blockDim.x + threadIdx.x;
  if (i >= n) return;
  float v = p[i];
  p[i] = (v > 0.0f) ? rsqrtf(v) : 0.0f;
}

// out[n][c] = sum_r b[r][c]   (b is [R, DOUT])
template <int DOUT>
__global__ void __launch_bounds__(256)
init_bias(float* __restrict__ out, const float* __restrict__ b, int nNodes) {
  size_t i = (size_t)blockIdx.x * blockDim.x + threadIdx.x;
  if (i >= (size_t)nNodes * DOUT) return;
  int c = (int)(i % DOUT);
  out[i] = b[c] + b[DOUT + c] + b[2 * DOUT + c] + b[3 * DOUT + c];
}

// ------------------------------------------------------------------ GEMM
// One wave (32 threads) computes a 16x64 tile of  (norm_s ⊙ h) @ W.
// A-frag 16x4 f32: lanes 0-15 -> K = {k,k+1}, lanes 16-31 -> K = {k+2,k+3}.
// B-frag 4x16 f32: mirrored layout, N = lane%16 from row-major W[din][DOUT].
// C/D v8f: VGPR i -> row i (lanes 0-15) / row i+8 (lanes 16-31).
template <int DOUT>
__global__ void __launch_bounds__(32)
rgcn_gemm_wmma(const float* __restrict__ h, const float* __restrict__ W,
               const float* __restrict__ norm_s, float* __restrict__ out,
               int nNodes) {
  const int lane   = threadIdx.x & 31;
  const int laneLo = lane & 15;
  const bool hiHalf = lane >= 16;
  const int row0 = blockIdx.x * 16;
  const int col0 = blockIdx.y * 64;

  int m = row0 + laneLo;
  if (m >= nNodes) m = nNodes - 1;          // clamp: WMMA requires full EXEC
  const float ns = norm_s[m];
  const float* __restrict__ hrow = h + (size_t)m * DFEAT + (hiHalf ? 2 : 0);
  const float* __restrict__ wcol =
      W + col0 + laneLo + (hiHalf ? (size_t)2 * DOUT : (size_t)0);

  v8f acc0 = {}, acc1 = {}, acc2 = {}, acc3 = {};
#pragma unroll 4
  for (int k = 0; k < DFEAT; k += 4) {
    v2f a;
    a.x = hrow[k] * ns;
    a.y = hrow[k + 1] * ns;
    const float* wk = wcol + (size_t)k * DOUT;
    v2f b0, b1, b2, b3;
    b0.x = wk[0];   b0.y = wk[DOUT];
    b1.x = wk[16];  b1.y = wk[DOUT + 16];
    b2.x = wk[32];  b2.y = wk[DOUT + 32];
    b3.x = wk[48];  b3.y = wk[DOUT + 48];
    acc0 = __builtin_amdgcn_wmma_f32_16x16x4_f32(false, a, false, b0, (short)0,
                                                 acc0, false, false);
    acc1 = __builtin_amdgcn_wmma_f32_16x16x4_f32(false, a, false, b1, (short)0,
                                                 acc1, false, false);
    acc2 = __builtin_amdgcn_wmma_f32_16x16x4_f32(false, a, false, b2, (short)0,
                                                 acc2, false, false);
    acc3 = __builtin_amdgcn_wmma_f32_16x16x4_f32(false, a, false, b3, (short)0,
                                                 acc3, false, false);
  }

  const int rbase = hiHalf ? 8 : 0;
  float* __restrict__ o = out + (size_t)row0 * DOUT + col0 + laneLo;
#pragma unroll
  for (int vg = 0; vg < 8; ++vg) {
    if (row0 + rbase + vg < nNodes) {
      size_t off = (size_t)(rbase + vg) * DOUT;
      o[off]      = acc0[vg];
      o[off + 16] = acc1[vg];
      o[off + 32] = acc2[vg];
      o[off + 48] = acc3[vg];
    }
  }
}

// ---------------------------------------------------------------- scatter
// out[dst[e]] += norm_d[dst[e]] * tmp[src[e]]   (float4 per thread)
template <int DOUT>
__global__ void __launch_bounds__(256)
rgcn_scatter(const float* __restrict__ tmp, const int* __restrict__ src,
             const int* __restrict__ dst, const float* __restrict__ norm_d,
             float* __restrict__ out, int nEdges) {
  constexpr int TPE = DOUT / 4;     // threads per edge
  constexpr int EPB = 256 / TPE;    // edges per block
  const int e = blockIdx.x * EPB + (int)(threadIdx.x / TPE);
  if (e >= nEdges) return;
  const int c = (int)(threadIdx.x % TPE) * 4;
  const int s = src[e];
  const int d = dst[e];
  const float nd = norm_d[d];
  const float4 v = *reinterpret_cast<const float4*>(tmp + (size_t)s * DOUT + c);
  float* o = out + (size_t)d * DOUT + c;
  atomAddF32(o + 0, v.x * nd);
  atomAddF32(o + 1, v.y * nd);
  atomAddF32(o + 2, v.z * nd);
  atomAddF32(o + 3, v.w * nd);
}

// ---------------------------------------------------------------- driver
extern "C" void kernel_launch(void* const* d_in, const int* in_sizes, int n_in,
                              void* d_out, int out_size, void* d_ws,
                              size_t ws_size, hipStream_t stream) {
  const float* x  = (const float*)d_in[0];
  const int* src  = (const int*)d_in[1];
  const int* dst  = (const int*)d_in[2];
  const float* Wl[6];
  const float* Bl[6];
  for (int i = 0; i < 6; ++i) {
    Wl[i] = (const float*)d_in[3 + 2 * i];
    Bl[i] = (const float*)d_in[4 + 2 * i];
  }

  const int n  = in_sizes[0] / DFEAT;   // nodes (100000)
  const int nE = in_sizes[1] / R_REL;   // edges per relation (400000)

  // workspace layout (floats): hA | hB | tmp | norm_s[R*n] | norm_d[R*n]
  float* ws    = (float*)d_ws;
  float* hA    = ws;
  float* hB    = hA + (size_t)n * DFEAT;
  float* tmp   = hB + (size_t)n * DFEAT;
  float* normS = tmp + (size_t)n * DFEAT;
  float* normD = normS + (size_t)R_REL * n;

  // --- degree -> norm (graph is layer-invariant: compute once per call) ---
  const size_t degTotal = (size_t)2 * R_REL * n;  // normS and normD contiguous
  fill_zero_f32<<<(unsigned)((degTotal + 255) / 256), 256, 0, stream>>>(
      normS, degTotal);
  const size_t cw = (size_t)R_REL * nE;
  count_degrees<<<(unsigned)((cw + 255) / 256), 256, 0, stream>>>(
      src, dst, normS, normD, n, nE);
  deg_to_norm<<<(unsigned)((degTotal + 255) / 256), 256, 0, stream>>>(
      normS, degTotal);

  // --- 6 layers: dims 128->128 (x5), 128->64 ---
  const float* cur = x;
  for (int L = 0; L < 6; ++L) {
    const bool last = (L == 5);
    float* next = last ? (float*)d_out : ((L & 1) ? hB : hA);

    if (last) {
      constexpr int DOUT = 64;
      init_bias<DOUT><<<(unsigned)(((size_t)n * DOUT + 255) / 256), 256, 0,
                        stream>>>(next, Bl[L], n);
      for (int r = 0; r < R_REL; ++r) {
        dim3 g((unsigned)((n + 15) / 16), DOUT / 64);
        rgcn_gemm_wmma<DOUT><<<g, 32, 0, stream>>>(
            cur, Wl[L] + (size_t)r * DFEAT * DOUT, normS + (size_t)r * n, tmp,
            n);
        constexpr int EPB = 256 / (DOUT / 4);
        rgcn_scatter<DOUT><<<(unsigned)((nE + EPB - 1) / EPB), 256, 0,
                             stream>>>(tmp, src + (size_t)r * nE,
                                       dst + (size_t)r * nE,
                                       normD + (size_t)r * n, next, nE);
      }
    } else {
      constexpr int DOUT = 128;
      init_bias<DOUT><<<(unsigned)(((size_t)n * DOUT + 255) / 256), 256, 0,
                        stream>>>(next, Bl[L], n);
      for (int r = 0; r < R_REL; ++r) {
        dim3 g((unsigned)((n + 15) / 16), DOUT / 64);
        rgcn_gemm_wmma<DOUT><<<g, 32, 0, stream>>>(
            cur, Wl[L] + (size_t)r * DFEAT * DOUT, normS + (size_t)r * n, tmp,
            n);
        constexpr int EPB = 256 / (DOUT / 4);
        rgcn_scatter<DOUT><<<(unsigned)((nE + EPB - 1) / EPB), 256, 0,
                             stream>>>(tmp, src + (size_t)r * nE,
                                       dst + (size_t)r * nE,
                                       normD + (size_t)r * n, next, nE);
      }
    }
    cur = next;
  }
}